// GraphConvolutionDiagLayer_68204080660515
// MI455X (gfx1250) — compile-verified
//
#include <hip/hip_runtime.h>

// GraphConvolutionDiagLayer: out = A_coo @ (x * diag(W))
//   x: [N,50] f32, W: [50] f32, edge_val: [E] f32, edge_row/col: [E] i32
// Memory-bound scatter/gather SpMM; working set ~60MB -> L2-resident on MI455X.

#define D_FEAT 50

constexpr int EDGES_PER_WAVE  = 16;
constexpr int WAVES_PER_BLOCK = 8;
constexpr int EDGES_PER_BLOCK = EDGES_PER_WAVE * WAVES_PER_BLOCK; // 128
constexpr int BLOCK_THREADS   = 256;

#if defined(__gfx1250__) && __has_builtin(__builtin_amdgcn_global_load_async_to_lds_b32)
#define USE_ASYNC_LDS 1
#else
#define USE_ASYNC_LDS 0
#endif

__global__ __launch_bounds__(BLOCK_THREADS)
void gc_spmm_kernel(const float* __restrict__ x,
                    const float* __restrict__ W,
                    const float* __restrict__ eval,
                    const int*   __restrict__ erow,
                    const int*   __restrict__ ecol,
                    float* __restrict__ out,
                    int n_edges) {
    __shared__ int   s_col[EDGES_PER_BLOCK];
    __shared__ int   s_row[EDGES_PER_BLOCK];
    __shared__ float s_val[EDGES_PER_BLOCK];
    __shared__ float s_W[64];

    const int tid       = threadIdx.x;
    const int blockBase = blockIdx.x * EDGES_PER_BLOCK;

    // ---- FIRST: stage this block's 128 edge records into LDS via the
    // async copy engine (ASYNCcnt path). Placed first so it leads the
    // kernel's program order.
    if (tid < EDGES_PER_BLOCK) {
        int e = blockBase + tid;
        if (e < n_edges) {
#if USE_ASYNC_LDS
            typedef __attribute__((address_space(1))) int* g32_t;
            typedef __attribute__((address_space(3))) int* l32_t;
            const int* eval_i = reinterpret_cast<const int*>(eval);
            __builtin_amdgcn_global_load_async_to_lds_b32(
                (g32_t)(ecol + e), (l32_t)(&s_col[tid]), 0, 0);
            __builtin_amdgcn_global_load_async_to_lds_b32(
                (g32_t)(erow + e), (l32_t)(&s_row[tid]), 0, 0);
            __builtin_amdgcn_global_load_async_to_lds_b32(
                (g32_t)(eval_i + e), (l32_t)(reinterpret_cast<int*>(&s_val[tid])), 0, 0);
#else
            s_col[tid] = ecol[e];
            s_row[tid] = erow[e];
            s_val[tid] = eval[e];
#endif
        }
    }

    // ---- Then stage the 50 diag weights (zero-padded to 64) with upper threads.
    if (tid >= 192) {
        int wi = tid - 192;   // 0..63
        s_W[wi] = (wi < D_FEAT) ? W[wi] : 0.0f;
    }

#if USE_ASYNC_LDS
#if __has_builtin(__builtin_amdgcn_s_wait_asynccnt)
    __builtin_amdgcn_s_wait_asynccnt(0);
#else
    asm volatile("s_wait_asynccnt 0x0" ::: "memory");
#endif
#endif
    __syncthreads();

    const int wave = tid >> 5;   // 0..7
    const int lane = tid & 31;   // 0..31

    // Per-lane diag weights for feature d = lane and d = lane+32.
    const float wA = s_W[lane];
    const float wB = s_W[lane + 32];   // zero-padded beyond 50 (unused lanes)

    const int waveBase = wave * EDGES_PER_WAVE;

    // Trip count for this wave (grid exactly covers E=1.6M, so normally 16).
    int kmax = n_edges - (blockBase + waveBase);
    if (kmax > EDGES_PER_WAVE) kmax = EDGES_PER_WAVE;

#pragma unroll 4
    for (int k = 0; k < kmax; ++k) {
        const int li = waveBase + k;

        const int   col = s_col[li];
        const int   row = s_row[li];
        const float val = s_val[li];

        const float* __restrict__ src = x   + (size_t)col * D_FEAT;
        float*       __restrict__ dst = out + (size_t)row * D_FEAT;

        // Features 0..31: all 32 lanes, coalesced 128B load + f32 atomics.
        const float a = src[lane];
        unsafeAtomicAdd(dst + lane, val * wA * a);

        // Features 32..49: lanes 0..17.
        if (lane < (D_FEAT - 32)) {
            const float b = src[lane + 32];
            unsafeAtomicAdd(dst + lane + 32, val * wB * b);
        }
    }
}

__global__ __launch_bounds__(BLOCK_THREADS)
void gc_zero_kernel(float4* __restrict__ out, int n4) {
    int i = blockIdx.x * blockDim.x + threadIdx.x;
    int stride = gridDim.x * blockDim.x;
    for (; i < n4; i += stride) {
        out[i] = make_float4(0.f, 0.f, 0.f, 0.f);
    }
}

extern "C" void kernel_launch(void* const* d_in, const int* in_sizes, int n_in,
                              void* d_out, int out_size, void* d_ws, size_t ws_size,
                              hipStream_t stream) {
    const float* x    = (const float*)d_in[0];   // [N*50]
    const float* W    = (const float*)d_in[1];   // [50]
    const float* eval = (const float*)d_in[2];   // [E]
    const int*   erow = (const int*)  d_in[3];   // [E]
    const int*   ecol = (const int*)  d_in[4];   // [E]
    float*       out  = (float*)d_out;           // [N*50]

    const int n_edges = in_sizes[2];

    // 1) zero the output (harness poisons it with 0xAA).
    {
        const int n4 = out_size / 4;                 // 5,000,000 / 4
        const int blocks = (n4 + BLOCK_THREADS - 1) / BLOCK_THREADS;
        gc_zero_kernel<<<blocks, BLOCK_THREADS, 0, stream>>>((float4*)out, n4);
    }

    // 2) fused gather-scale-scatter SpMM.
    {
        const int blocks = (n_edges + EDGES_PER_BLOCK - 1) / EDGES_PER_BLOCK; // 12,500
        gc_spmm_kernel<<<blocks, BLOCK_THREADS, 0, stream>>>(
            x, W, eval, erow, ecol, out, n_edges);
    }
}